// DentalVolumeProcessor_17411797418423
// MI455X (gfx1250) — compile-verified
//
#include <hip/hip_runtime.h>

typedef __attribute__((ext_vector_type(2))) float v2f;
typedef __attribute__((ext_vector_type(8))) float v8f;

#define G1 0.6065306597126334f   // exp(-0.5)
#define G2 0.1353352832366127f   // exp(-2)

__device__ __forceinline__ float gof(int m) {
    int a = m < 0 ? -m : m;
    return a == 0 ? 1.0f : (a == 1 ? G1 : (a == 2 ? G2 : 0.0f));
}

// K[d,j] = sum over pool window e in {d-1,d,d+1} (clipped to [0,128)) of gauss(e-j)
__device__ __forceinline__ float kel(int d, int j) {
    float v = gof(d - j);
    if (d >= 1)   v += gof(d - 1 - j);
    if (d <= 126) v += gof(d + 1 - j);
    return v;
}

#define OH_STRIDE 72   // 2*72 % 64 == 16 -> conflict-free B-fragment loads

__global__ __launch_bounds__(256)
void dental_vol_kernel(const float* __restrict__ depth,
                       const float* __restrict__ xray,
                       float* __restrict__ out)
{
    __shared__ float OH[128 * OH_STRIDE];   // depth-index histogram [j][w_local]
    __shared__ float xr_s[3 * OH_STRIDE];   // staged downsampled x_ray (3 rows, halo cols)
    __shared__ int   di_s[3 * OH_STRIDE];   // staged depth indices

    const int tid   = threadIdx.x;
    const int wBase = blockIdx.x * 64;      // 0 or 64
    const int h     = blockIdx.y;           // 0..127
    const int b     = blockIdx.z;           // 0..15

    const float* dm = depth + (size_t)b * 256 * 256;
    const float* xm = xray  + (size_t)b * 256 * 256;

    const float w4[4] = {0.125f, 0.375f, 0.375f, 0.125f};

    // ---- Stage: antialiased 256->128 downsample of (depth, xray) for 3 rows + w halo ----
    for (int idx = tid; idx < 3 * 66; idx += 256) {
        int r  = idx / 66;            // 0..2  -> row h-1+r
        int c  = idx % 66;            // 0..65 -> col wBase-1+c
        int hh = h - 1 + r;
        int ww = wBase - 1 + c;
        float xv = 0.0f;
        int   jv = 0;
        if (hh >= 0 && hh < 128 && ww >= 0 && ww < 128) {
            float accd = 0.f, accx = 0.f, sy = 0.f, sx = 0.f;
            #pragma unroll
            for (int ci = 0; ci < 4; ++ci) {
                int cc = 2 * ww - 1 + ci;
                if (cc >= 0 && cc < 256) sx += w4[ci];
            }
            #pragma unroll
            for (int ri = 0; ri < 4; ++ri) {
                int rr = 2 * hh - 1 + ri;
                if (rr < 0 || rr >= 256) continue;
                sy += w4[ri];
                const float* drow = dm + rr * 256;
                const float* xrow = xm + rr * 256;
                float rd = 0.f, rx = 0.f;
                #pragma unroll
                for (int ci = 0; ci < 4; ++ci) {
                    int cc = 2 * ww - 1 + ci;
                    if (cc < 0 || cc >= 256) continue;
                    rd += w4[ci] * drow[cc];
                    rx += w4[ci] * xrow[cc];
                }
                accd += w4[ri] * rd;
                accx += w4[ri] * rx;
            }
            float inv = 1.0f / (sx * sy);
            float dr  = accd * inv;
            xv        = accx * inv;
            int di = (int)(dr * 127.0f);          // truncation, matches .astype(int32)
            jv = di < 0 ? 0 : (di > 127 ? 127 : di);
        }
        xr_s[r * OH_STRIDE + c] = xv;
        di_s[r * OH_STRIDE + c] = jv;
    }

    // ---- Zero histogram ----
    for (int i = tid; i < 128 * OH_STRIDE; i += 256) OH[i] = 0.0f;
    __syncthreads();

    // ---- Scatter: OH[j][w] = sum over 3x3 spatial neighbors of xr where d_idx==j ----
    if (tid < 64) {
        int w = tid;
        #pragma unroll
        for (int dh = 0; dh < 3; ++dh) {
            #pragma unroll
            for (int dw = 0; dw < 3; ++dw) {
                int c = w + dw;   // stage col of (w + dw - 1) is (w + dw - 1) + 1
                float xv = xr_s[dh * OH_STRIDE + c];
                int   j  = di_s[dh * OH_STRIDE + c];
                atomicAdd(&OH[j * OH_STRIDE + w], xv);   // ds_add_f32, column-private
            }
        }
    }
    __syncthreads();

    // ---- Banded matmul: OUT(16I..16I+15, cols) = sum_J K(I,J) x OH(J, cols) via WMMA ----
    const int wv    = tid >> 5;       // wave id 0..7 == d-tile I
    const int lane  = tid & 31;
    const int l15   = lane & 15;
    const int h2    = lane >> 4;      // lane half
    const int I     = wv;
    const int dBase = I * 16;

    // A fragments of constant K: 16x4 f32 layout -> a.x = K[d, j0], a.y = K[d, j0+1],
    // with j0 = tile_j_base + 2*h2 (VGPR0 holds K={0,2}, VGPR1 holds K={1,3}).
    v2f af[3][4];
    #pragma unroll
    for (int jj = 0; jj < 3; ++jj) {
        int J = I - 1 + jj;
        #pragma unroll
        for (int c = 0; c < 4; ++c) {
            int j0 = J * 16 + c * 4 + 2 * h2;
            int d  = dBase + l15;
            v2f a;
            a.x = kel(d, j0);
            a.y = kel(d, j0 + 1);
            af[jj][c] = a;
        }
    }
    const int jlo = (I == 0) ? 1 : 0;   // wave-uniform: EXEC stays all-ones
    const int jhi = (I == 7) ? 1 : 2;

    float* outp = out + (size_t)b * (128 * 128 * 128);
    const float scale = 1.0f / 27.0f;

    for (int Jc = 0; Jc < 4; ++Jc) {
        int cb = Jc * 16;
        v8f acc = {};
        #pragma unroll
        for (int jj = 0; jj < 3; ++jj) {
            if (jj < jlo || jj > jhi) continue;   // wave-uniform branch
            int J = I - 1 + jj;
            #pragma unroll
            for (int c = 0; c < 4; ++c) {
                int j0 = J * 16 + c * 4 + 2 * h2;
                v2f bf;
                bf.x = OH[(j0    ) * OH_STRIDE + cb + l15];
                bf.y = OH[(j0 + 1) * OH_STRIDE + cb + l15];
                acc = __builtin_amdgcn_wmma_f32_16x16x4_f32(
                        /*neg_a=*/false, af[jj][c],
                        /*neg_b=*/false, bf,
                        /*c_mod=*/(short)0, acc,
                        /*reuse_a=*/false, /*reuse_b=*/false);
            }
        }
        // D layout: VGPR v -> row d = dBase + v + 8*h2, col = cb + l15
        int w = wBase + cb + l15;
        #pragma unroll
        for (int v = 0; v < 8; ++v) {
            int d = dBase + v + 8 * h2;
            outp[(size_t)(d * 16384 + h * 128 + w)] = acc[v] * scale;
        }
    }
}

extern "C" void kernel_launch(void* const* d_in, const int* in_sizes, int n_in,
                              void* d_out, int out_size, void* d_ws, size_t ws_size,
                              hipStream_t stream) {
    const float* depth = (const float*)d_in[0];
    const float* xray  = (const float*)d_in[1];
    float* out = (float*)d_out;
    dim3 grid(2, 128, 16);   // (w-half, h, batch)
    dim3 block(256);
    hipLaunchKernelGGL(dental_vol_kernel, grid, block, 0, stream, depth, xray, out);
}